// SCALayer_54958401520060
// MI455X (gfx1250) — compile-verified
//
#include <hip/hip_runtime.h>
#include <math.h>

typedef __attribute__((ext_vector_type(2))) float v2f;
typedef __attribute__((ext_vector_type(8))) float v8f;

#define CCH 128           // channel count (C_IN == C_OUT == 128)
#define LDS_STRIDE 132    // 16x128 x-tile padded: 132 % 64 = 4 banks -> conflict-free

__device__ __forceinline__ float sigmoidf_(float x) {
  return 1.0f / (1.0f + __expf(-x));
}

// ---------------------------------------------------------------- zero scratch
__global__ void sca_zero_f32(float* __restrict__ p, long long n) {
  long long i = (long long)blockIdx.x * blockDim.x + threadIdx.x;
  long long stride = (long long)gridDim.x * blockDim.x;
  for (; i < n; i += stride) p[i] = 0.0f;
}

// ------------------------------------------------- xm = x @ W (fp32 WMMA GEMM)
// block = 256 threads (8 wave32). Block computes rows [row0, row0+16) x all 128
// cols; wave w owns cols [16w, 16w+16). A/B/C per-lane layouts follow the CDNA5
// ISA 16x4 / 4x16 / 16x16 f32 VGPR maps.
__global__ void sca_gemm_xw_wmma(const float* __restrict__ x,
                                 const float* __restrict__ W,
                                 float* __restrict__ xm, int n) {
  __shared__ float xs[16 * LDS_STRIDE];
  const int row0 = blockIdx.x * 16;
  const int tid = threadIdx.x;

  // cooperative load: 16x128 floats = 512 float4s, 2 per thread
#pragma unroll
  for (int j = 0; j < 2; ++j) {
    int flat = tid * 2 + j;        // 0..511
    int m = flat >> 5;             // 0..15
    int k = (flat & 31) * 4;       // 0..124
    float4 v = make_float4(0.f, 0.f, 0.f, 0.f);
    int row = row0 + m;
    if (row < n) v = *(const float4*)(x + (size_t)row * CCH + k);
    *(float4*)(xs + m * LDS_STRIDE + k) = v;
  }
  __syncthreads();

  const int wave = tid >> 5;
  const int lane = tid & 31;
  const int half = lane >> 4;      // 0: K pair {0,1}; 1: K pair {2,3}
  const int l15  = lane & 15;
  const int koff = half * 2;
  const int col0 = wave * 16;

  v8f acc = {};
  for (int k = 0; k < CCH; k += 4) {
    v2f a, b;
    // A fragment (16x4): lanes 0-15 rows M=0..15 K={k,k+1}; lanes 16-31 K={k+2,k+3}
    const float* xp = xs + l15 * LDS_STRIDE + k + koff;
    a[0] = xp[0];
    a[1] = xp[1];
    // B fragment (4x16): lanes 0-15 cols N, K rows {k,k+1}; lanes 16-31 {k+2,k+3}
    const float* wp = W + (size_t)(k + koff) * CCH + col0 + l15;
    b[0] = wp[0];
    b[1] = wp[CCH];
    acc = __builtin_amdgcn_wmma_f32_16x16x4_f32(
        /*neg_a=*/false, a, /*neg_b=*/false, b,
        /*c_mod=*/(short)0, acc, /*reuse_a=*/false, /*reuse_b=*/false);
  }

  // C/D 16x16 f32: VGPR r -> M = r + 8*half, N = l15
  const int col = col0 + l15;
#pragma unroll
  for (int r = 0; r < 8; ++r) {
    int row = row0 + r + half * 8;
    if (row < n) xm[(size_t)row * CCH + col] = acc[r];
  }
}

// -------------------------------------- h[row] += val * xm[col]  (COO scatter)
// one wave32 per nnz; one float4 per lane covers the 128-channel row;
// unsafeAtomicAdd -> global_atomic_add_f32 (resolves in L2: xm+h fit in 192MB)
__global__ void sca_scatter(const float* __restrict__ xm,
                            const int* __restrict__ rows,
                            const int* __restrict__ cols,
                            const float* __restrict__ vals,
                            float* __restrict__ h, int nnz) {
  int wv = (int)(((long long)blockIdx.x * blockDim.x + threadIdx.x) >> 5);
  int lane = threadIdx.x & 31;
  if (wv >= nnz) return;
  int r = rows[wv];
  int c = cols[wv];
  float v = vals[wv];
  float4 xvec = ((const float4*)(xm + (size_t)c * CCH))[lane];
  float* dst = h + (size_t)r * CCH + lane * 4;
  unsafeAtomicAdd(dst + 0, v * xvec.x);
  unsafeAtomicAdd(dst + 1, v * xvec.y);
  unsafeAtomicAdd(dst + 2, v * xvec.z);
  unsafeAtomicAdd(dst + 3, v * xvec.w);
}

// -------------------------------------------- wsum[c] = sum_r h[r][c] (128 ch)
__global__ void sca_colsum(const float* __restrict__ h,
                           float* __restrict__ wsum, int n_rows) {
  int c = threadIdx.x & (CCH - 1);
  int sub = threadIdx.x >> 7;    // 0..1: two rows per block-iteration
  float acc = 0.0f;
  for (int r = blockIdx.x * 2 + sub; r < n_rows; r += gridDim.x * 2)
    acc += h[(size_t)r * CCH + c];
  __shared__ float red[256];
  red[threadIdx.x] = acc;
  __syncthreads();
  if (sub == 0) unsafeAtomicAdd(&wsum[c], red[c] + red[c + CCH]);
}

// ---- fused: reweight(h1), reweight(h2), out = sigmoid((s1*h1 + s2*h2) * 0.5)
// one wave32 per output row; per-row dot via wave shuffle reduction
__global__ void sca_finalize(const float* __restrict__ h1,
                             const float* __restrict__ h2,
                             const float* __restrict__ w1s,
                             const float* __restrict__ w2s,
                             float* __restrict__ out, int n) {
  int gw = (int)(((long long)blockIdx.x * blockDim.x + threadIdx.x) >> 5);
  int lane = threadIdx.x & 31;
  if (gw >= n) return;

  const float4 a  = ((const float4*)(h1 + (size_t)gw * CCH))[lane];
  const float4 b  = ((const float4*)(h2 + (size_t)gw * CCH))[lane];
  const float4 wa = ((const float4*)w1s)[lane];
  const float4 wb = ((const float4*)w2s)[lane];

  // w = relu(sigmoid(colsum)); sigmoid > 0 so relu is identity (kept for fidelity)
  float4 sa, sb;
  sa.x = fmaxf(sigmoidf_(wa.x), 0.f); sa.y = fmaxf(sigmoidf_(wa.y), 0.f);
  sa.z = fmaxf(sigmoidf_(wa.z), 0.f); sa.w = fmaxf(sigmoidf_(wa.w), 0.f);
  sb.x = fmaxf(sigmoidf_(wb.x), 0.f); sb.y = fmaxf(sigmoidf_(wb.y), 0.f);
  sb.z = fmaxf(sigmoidf_(wb.z), 0.f); sb.w = fmaxf(sigmoidf_(wb.w), 0.f);

  float d1 = sa.x * a.x + sa.y * a.y + sa.z * a.z + sa.w * a.w;
  float d2 = sb.x * b.x + sb.y * b.y + sb.z * b.z + sb.w * b.w;
#pragma unroll
  for (int m = 16; m > 0; m >>= 1) {
    d1 += __shfl_xor(d1, m, 32);
    d2 += __shfl_xor(d2, m, 32);
  }
  float s1 = sigmoidf_(d1);
  float s2 = sigmoidf_(d2);

  float4 o;
  o.x = sigmoidf_((s1 * a.x + s2 * b.x) * 0.5f);
  o.y = sigmoidf_((s1 * a.y + s2 * b.y) * 0.5f);
  o.z = sigmoidf_((s1 * a.z + s2 * b.z) * 0.5f);
  o.w = sigmoidf_((s1 * a.w + s2 * b.w) * 0.5f);
  ((float4*)(out + (size_t)gw * CCH))[lane] = o;
}

extern "C" void kernel_launch(void* const* d_in, const int* in_sizes, int n_in,
                              void* d_out, int out_size, void* d_ws, size_t ws_size,
                              hipStream_t stream) {
  const float* x1  = (const float*)d_in[0];
  const float* x2  = (const float*)d_in[1];
  const int*   n1r = (const int*)d_in[2];
  const int*   n1c = (const int*)d_in[3];
  const float* n1v = (const float*)d_in[4];
  const int*   n2r = (const int*)d_in[5];
  const int*   n2c = (const int*)d_in[6];
  const float* n2v = (const float*)d_in[7];
  const float* W1  = (const float*)d_in[8];
  const float* W2  = (const float*)d_in[9];

  const int nk   = in_sizes[0] / CCH;   // N_K
  const int nl   = in_sizes[1] / CCH;   // N_L
  const int nnz1 = in_sizes[2];
  const int nnz2 = in_sizes[5];
  const int nm   = out_size / CCH;      // N_M

  // workspace layout (floats): [xm1 | xm2 | h1 | h2 | w1sum(128) | w2sum(128)]
  float* ws  = (float*)d_ws;
  float* xm1 = ws;
  float* xm2 = xm1 + (size_t)nk * CCH;
  float* h1  = xm2 + (size_t)nl * CCH;
  float* h2  = h1 + (size_t)nm * CCH;
  float* w1s = h2 + (size_t)nm * CCH;
  float* w2s = w1s + CCH;

  // 1) zero accumulators (h1, h2, w1s, w2s are contiguous) — every call
  long long zn = (long long)nm * CCH * 2 + 2 * CCH;
  sca_zero_f32<<<1024, 256, 0, stream>>>(h1, zn);

  // 2) dense GEMMs via fp32 WMMA
  sca_gemm_xw_wmma<<<(nk + 15) / 16, 256, 0, stream>>>(x1, W1, xm1, nk);
  sca_gemm_xw_wmma<<<(nl + 15) / 16, 256, 0, stream>>>(x2, W2, xm2, nl);

  // 3) COO gather/scale/scatter (one wave per nnz)
  {
    long long thr1 = (long long)nnz1 * 32;
    long long thr2 = (long long)nnz2 * 32;
    sca_scatter<<<(int)((thr1 + 255) / 256), 256, 0, stream>>>(xm1, n1r, n1c, n1v, h1, nnz1);
    sca_scatter<<<(int)((thr2 + 255) / 256), 256, 0, stream>>>(xm2, n2r, n2c, n2v, h2, nnz2);
  }

  // 4) column sums for reweighting
  sca_colsum<<<512, 256, 0, stream>>>(h1, w1s, nm);
  sca_colsum<<<512, 256, 0, stream>>>(h2, w2s, nm);

  // 5) fused reweight + merge + sigmoid
  {
    long long thr = (long long)nm * 32;
    sca_finalize<<<(int)((thr + 255) / 256), 256, 0, stream>>>(h1, h2, w1s, w2s,
                                                               (float*)d_out, nm);
  }
}